// DFNet_37108517437896
// MI455X (gfx1250) — compile-verified
//
#include <hip/hip_runtime.h>
#include <hip/hip_bf16.h>

#define SUBSTEPS 100

// Single-lane serial integrator for the nonlinear 2-state Euler recurrence.
// All state is wave-uniform, so clang lowers this to the gfx1250 scalar-float
// pipeline (s_fmac_f32 / s_fmaak_f32 / v_s_rcp_f32) with ~1-cycle dependent
// latency and SALU/VALU co-execution. Per-substep cost after algebraic
// refactoring: 8 float ops, critical cycle ~3.5 dependent ops/step.
//
// Refactored update (mathematically identical to the reference):
//   d  = r*r + b2^2                       -> fma
//   q  = 1/d                              -> v_s_rcp_f32 + 1 Newton refine
//   I' = k5*I + c4 - c4*b2^2*q            (uses c4*rs/d == c4 - c4*b2^2/d)
//   r' = r*((1-c2) - c3*I) + c1
__global__ __launch_bounds__(32, 1)
void DFNet_scan_kernel(const float* __restrict__ x,
                       const float* __restrict__ pa0,
                       const float* __restrict__ pa1,
                       const float* __restrict__ pa2,
                       const float* __restrict__ pb1,
                       const float* __restrict__ pb2,
                       const float* __restrict__ pb3,
                       const float* __restrict__ pI0,
                       float* __restrict__ out,
                       int n)   // n == 8192 output samples
{
    const float dt2 = 0.3f;                  // 2 * DELTA_T
    const float a0  = pa0[0];
    const float a1  = pa1[0];
    const float a2  = pa2[0];
    const float b1  = pb1[0];
    const float b2  = pb2[0];
    const float b3  = pb3[0];

    const float c1    = dt2 * a0;            // additive r term
    const float tkA   = 1.0f - dt2 * a1;     // (1 - c2)
    const float c3    = dt2 * a2;            // coeff of r*I
    const float c4    = dt2 * b1;            // dt2*b1
    const float k5    = 1.0f - dt2 * b3;     // I decay factor
    const float b2sq  = b2 * b2;
    const float nc4b2 = -(c4 * b2sq);        // -c4*b2^2  (coeff of q)

    float r = x[0];
    float I = pI0[0];

    out[0] = r;                              // G[0] = x[0]

    for (int o = 1; o < n; ++o) {
#pragma unroll
        for (int s = 0; s < SUBSTEPS; ++s) {
            // All reads use pre-update (r, I), matching the reference order.
            float d = __builtin_fmaf(r, r, b2sq);
            float q = __builtin_amdgcn_rcpf(d);            // scalar rcp
            q = q * __builtin_fmaf(-d, q, 2.0f);           // 1x Newton refine

            float t    = __builtin_fmaf(-c3, I, tkA);      // (1-c2) - c3*I
            float rnew = __builtin_fmaf(r, t, c1);         // r*t + c1
            float iacc = __builtin_fmaf(k5, I, c4);        // k5*I + c4  (fmaak)
            float inew = __builtin_fmaf(nc4b2, q, iacc);   // - c4*b2^2*q

            r = rnew;
            I = inew;
        }
        out[o] = r;                          // G[o] = r after o*100 substeps
    }
    // Reference runs 99 extra substeps past the last sample; they affect no
    // output element, so they are skipped.
}

extern "C" void kernel_launch(void* const* d_in, const int* in_sizes, int n_in,
                              void* d_out, int out_size, void* d_ws, size_t ws_size,
                              hipStream_t stream) {
    (void)n_in; (void)d_ws; (void)ws_size; (void)in_sizes;

    const float* x  = (const float*)d_in[0];
    const float* a0 = (const float*)d_in[1];
    const float* a1 = (const float*)d_in[2];
    const float* a2 = (const float*)d_in[3];
    const float* b1 = (const float*)d_in[4];
    const float* b2 = (const float*)d_in[5];
    const float* b3 = (const float*)d_in[6];
    const float* I0 = (const float*)d_in[7];
    float* out = (float*)d_out;

    const int n = out_size;                  // 8192

    DFNet_scan_kernel<<<dim3(1), dim3(1), 0, stream>>>(
        x, a0, a1, a2, b1, b2, b3, I0, out, n);
}